// GCNN_69990787055800
// MI455X (gfx1250) — compile-verified
//
#include <hip/hip_runtime.h>
#include <math.h>

// ---------------------------------------------------------------------------
// GCNN rollout on MI455X (gfx1250), fp32 WMMA path (V_WMMA_F32_16X16X4_F32).
//  - one workgroup per batch (64 WGPs), whole T=64 recurrence in-kernel
//  - recurrent state in LDS (~99KB of 320KB/WGP), S streamed from L2 (4MB)
//  - two row-tiles per wave share B operands (2 independent WMMA chains)
//  - value-masking (mul by lane mask) instead of exec-predicated ds_loads
//  - x0/H2 staged via GLOBAL_LOAD_ASYNC_TO_LDS (ASYNCcnt) at startup
//  - trajectory written with non-temporal stores (keep S/H resident in L2)
// ---------------------------------------------------------------------------

typedef __attribute__((ext_vector_type(2))) float v2f;
typedef __attribute__((ext_vector_type(8))) float v8f;

#define N_NODES   1024
#define P_DIM     4
#define F_DIM     32
#define Q_DIM     4
#define T_STEPS   64
#define B_BATCH   64
#define DT_STEP   0.1f
#define THREADS   256
#define WAVES     (THREADS / 32)
#define TILES     (N_NODES / 16)       // 64 row tiles of 16
#define SX_STRIDE 16                   // (K+1)*p columns of diffusion taps

// LDS float counts (with padding so B fetches are always in-bounds)
#define SX_SZ   (N_NODES * SX_STRIDE)          // 16384
#define WBUF_SZ (N_NODES * Q_DIM + 16)         // 4112 (padded)
#define WSCR_SZ (WAVES * 16 * F_DIM)           // 4096 (Z scratch / stage buf)
#define H2_SZ   (F_DIM * Q_DIM + 16)           // 144  (padded)
#define LDS_FLOATS (SX_SZ + WBUF_SZ + WSCR_SZ + H2_SZ)

__device__ __forceinline__ v8f wmma_f32(v2f a, v2f b, v8f c) {
    // D = A(16x4,f32) * B(4x16,f32) + C(16x16,f32)
    return __builtin_amdgcn_wmma_f32_16x16x4_f32(false, a, false, b,
                                                 (short)0, c, false, false);
}

__device__ __forceinline__ void wait_asynccnt0() {
#if __has_builtin(__builtin_amdgcn_s_wait_asynccnt)
    __builtin_amdgcn_s_wait_asynccnt(0);
#else
    asm volatile("s_wait_asynccnt 0x0" ::: "memory");
#endif
}

// dst[0:N,0:4] = S @ src[0:N,0:4].  Two row tiles per wave share B loads.
// A lane layout (16x4 f32): lane<16 -> rows M=lane, K={0,1}; lane>=16 -> K={2,3}
__device__ __forceinline__ void s_pass(const float* __restrict__ Sg,
                                       const float* src, int sstride,
                                       float* dst, int dstride,
                                       int wave, int laneM, int half,
                                       float mfac) {
    for (int tt = wave; tt < TILES; tt += 2 * WAVES) {
        const int m0 = tt * 16;
        const int m1 = (tt + WAVES) * 16;
        v8f acc0 = {0.f,0.f,0.f,0.f,0.f,0.f,0.f,0.f};
        v8f acc1 = {0.f,0.f,0.f,0.f,0.f,0.f,0.f,0.f};
        const float* arow0 = Sg + (size_t)(m0 + laneM) * N_NODES + 2 * half;
        const float* arow1 = Sg + (size_t)(m1 + laneM) * N_NODES + 2 * half;
        for (int k0 = 0; k0 < N_NODES; k0 += 32) {
            __builtin_prefetch(arow0 + k0 + 512, 0, 1);   // global_prefetch_b8
            __builtin_prefetch(arow1 + k0 + 512, 0, 1);
            #pragma unroll
            for (int kk = 0; kk < 32; kk += 4) {
                const int k = k0 + kk;
                // B depends only on k: shared by both tiles; value-masked
                float b0v = src[(k + 2 * half) * sstride + laneM];
                float b1v = src[(k + 2 * half + 1) * sstride + laneM];
                v2f b; b.x = b0v * mfac; b.y = b1v * mfac;
                v2f a0; a0.x = arow0[k]; a0.y = arow0[k + 1];
                v2f a1; a1.x = arow1[k]; a1.y = arow1[k + 1];
                acc0 = wmma_f32(a0, b, acc0);
                acc1 = wmma_f32(a1, b, acc1);
            }
        }
        if (laneM < P_DIM) {
            #pragma unroll
            for (int i = 0; i < 8; ++i) {
                dst[(m0 + i + 8 * half) * dstride + laneM] = acc0[i];
                dst[(m1 + i + 8 * half) * dstride + laneM] = acc1[i];
            }
        }
    }
}

extern __shared__ float smem[];

__global__ __launch_bounds__(THREADS, 1)
void gcnn_rollout(const float* __restrict__ x0,
                  const float* __restrict__ Sg,
                  const float* __restrict__ H1g,   // (4,4,32) == flat (16,32)
                  const float* __restrict__ H2g,   // (1,32,4)
                  float* __restrict__ xtraj,       // (B,T+1,N,p)
                  float* __restrict__ utraj) {     // (B,T,N,q)
    float* sx   = smem;                  // [N][16] diffusion taps: S^k x
    float* wbuf = sx + SX_SZ;            // [N][4] (+pad)  W = Z @ H2
    float* wscr = wbuf + WBUF_SZ;        // [WAVES][16][32] Z scratch / stage
    float* h2s  = wscr + WSCR_SZ;        // [32][4] (+pad)

    const int tid   = threadIdx.x;
    const int wave  = tid >> 5;
    const int lane  = tid & 31;
    const int laneM = lane & 15;
    const int half  = lane >> 4;
    const int b     = blockIdx.x;
    const float mP  = (laneM < P_DIM) ? 1.f : 0.f;   // p/q column lane mask

    // Zero LDS once (taps cols 4..15 + all padding).
    for (int idx = tid; idx < LDS_FLOATS; idx += THREADS) smem[idx] = 0.f;
    __syncthreads();

    // ---- async-stage x0 slab (16KB) into wscr and H2 into h2s (ASYNCcnt) ----
    {
        const float* x0b = x0 + (size_t)b * N_NODES * P_DIM;
        const unsigned wscrOff = (unsigned)((const char*)wscr - (const char*)smem);
        for (int idx = tid * 4; idx < N_NODES * P_DIM; idx += THREADS * 4) {
            unsigned lo = wscrOff + (unsigned)idx * 4u;
            const float* gp = x0b + idx;
            asm volatile("global_load_async_to_lds_b128 %0, %1, off"
                         :: "v"(lo), "v"(gp) : "memory");
        }
        if (tid < F_DIM * Q_DIM) {
            unsigned lo = (unsigned)((const char*)h2s - (const char*)smem) +
                          (unsigned)tid * 4u;
            const float* gp = H2g + tid;
            asm volatile("global_load_async_to_lds_b32 %0, %1, off"
                         :: "v"(lo), "v"(gp) : "memory");
        }
        wait_asynccnt0();
        __syncthreads();
        // Rearrange contiguous [N][4] slab into sx columns 0..3, emit x_traj[b][0].
        for (int idx = tid; idx < N_NODES * P_DIM; idx += THREADS) {
            float v = wscr[idx];
            sx[(idx >> 2) * SX_STRIDE + (idx & 3)] = v;
            __builtin_nontemporal_store(
                v, xtraj + (size_t)b * (T_STEPS + 1) * N_NODES * P_DIM + idx);
        }
        __syncthreads();
    }

    float* myscr = wscr + wave * 16 * F_DIM;

    for (int t = 0; t < T_STEPS; ++t) {
        // ---- diffusion taps: columns 4k..4k+3 of sx hold S^k x ----
        s_pass(Sg, sx + 0, SX_STRIDE, sx + 4,  SX_STRIDE, wave, laneM, half, mP);
        __syncthreads();
        s_pass(Sg, sx + 4, SX_STRIDE, sx + 8,  SX_STRIDE, wave, laneM, half, mP);
        __syncthreads();
        s_pass(Sg, sx + 8, SX_STRIDE, sx + 12, SX_STRIDE, wave, laneM, half, mP);
        __syncthreads();

        // ---- fused per-tile: Zpre = sx @ H1 (K=16), tanh, W = Z @ H2 (K=32)
        for (int tt = wave; tt < TILES; tt += WAVES) {
            const int m = tt * 16;
            v8f z0 = {0.f,0.f,0.f,0.f,0.f,0.f,0.f,0.f};
            v8f z1 = {0.f,0.f,0.f,0.f,0.f,0.f,0.f,0.f};
            #pragma unroll
            for (int kc = 0; kc < SX_STRIDE; kc += 4) {
                const float* ap = sx + (m + laneM) * SX_STRIDE + kc + 2 * half;
                v2f a; a.x = ap[0]; a.y = ap[1];
                const float* bp = H1g + (kc + 2 * half) * F_DIM + laneM;
                v2f bA; bA.x = bp[0];  bA.y = bp[F_DIM];
                v2f bB; bB.x = bp[16]; bB.y = bp[F_DIM + 16];
                z0 = wmma_f32(a, bA, z0);
                z1 = wmma_f32(a, bB, z1);
            }
            #pragma unroll
            for (int i = 0; i < 8; ++i) {
                const int r = i + 8 * half;
                myscr[r * F_DIM + laneM]      = tanhf(z0[i]);
                myscr[r * F_DIM + 16 + laneM] = tanhf(z1[i]);
            }
            // W tile (K = 32 chained in 8 WMMAs); B cols >= q value-masked
            v8f w = {0.f,0.f,0.f,0.f,0.f,0.f,0.f,0.f};
            #pragma unroll
            for (int kc = 0; kc < F_DIM; kc += 4) {
                const float* ap2 = myscr + laneM * F_DIM + kc + 2 * half;
                v2f a2; a2.x = ap2[0]; a2.y = ap2[1];
                float b0v = h2s[(kc + 2 * half) * Q_DIM + laneM];
                float b1v = h2s[(kc + 2 * half + 1) * Q_DIM + laneM];
                v2f b2; b2.x = b0v * mP; b2.y = b1v * mP;
                w = wmma_f32(a2, b2, w);
            }
            if (laneM < Q_DIM) {
                #pragma unroll
                for (int i = 0; i < 8; ++i)
                    wbuf[(m + i + 8 * half) * Q_DIM + laneM] = w[i];
            }
        }
        __syncthreads();

        // ---- U = S @ W (two tiles/wave), x += DT*U, emit trajectory ----
        for (int tt = wave; tt < TILES; tt += 2 * WAVES) {
            const int m0 = tt * 16;
            const int m1 = (tt + WAVES) * 16;
            v8f acc0 = {0.f,0.f,0.f,0.f,0.f,0.f,0.f,0.f};
            v8f acc1 = {0.f,0.f,0.f,0.f,0.f,0.f,0.f,0.f};
            const float* arow0 = Sg + (size_t)(m0 + laneM) * N_NODES + 2 * half;
            const float* arow1 = Sg + (size_t)(m1 + laneM) * N_NODES + 2 * half;
            for (int k0 = 0; k0 < N_NODES; k0 += 32) {
                __builtin_prefetch(arow0 + k0 + 512, 0, 1);
                __builtin_prefetch(arow1 + k0 + 512, 0, 1);
                #pragma unroll
                for (int kk = 0; kk < 32; kk += 4) {
                    const int k = k0 + kk;
                    float b0v = wbuf[(k + 2 * half) * Q_DIM + laneM];
                    float b1v = wbuf[(k + 2 * half + 1) * Q_DIM + laneM];
                    v2f bq; bq.x = b0v * mP; bq.y = b1v * mP;
                    v2f a0; a0.x = arow0[k]; a0.y = arow0[k + 1];
                    v2f a1; a1.x = arow1[k]; a1.y = arow1[k + 1];
                    acc0 = wmma_f32(a0, bq, acc0);
                    acc1 = wmma_f32(a1, bq, acc1);
                }
            }
            if (laneM < Q_DIM) {
                #pragma unroll
                for (int i = 0; i < 8; ++i) {
                    const int r0 = m0 + i + 8 * half;
                    const int r1 = m1 + i + 8 * half;
                    const float u0 = acc0[i], u1 = acc1[i];
                    const float xn0 = sx[r0 * SX_STRIDE + laneM] + DT_STEP * u0;
                    const float xn1 = sx[r1 * SX_STRIDE + laneM] + DT_STEP * u1;
                    sx[r0 * SX_STRIDE + laneM] = xn0;
                    sx[r1 * SX_STRIDE + laneM] = xn1;
                    float* xt = xtraj + (((size_t)b * (T_STEPS + 1) + (t + 1)) *
                                         N_NODES) * P_DIM + laneM;
                    float* ut = utraj + (((size_t)b * T_STEPS + t) *
                                         N_NODES) * Q_DIM + laneM;
                    __builtin_nontemporal_store(xn0, xt + r0 * P_DIM);
                    __builtin_nontemporal_store(xn1, xt + r1 * P_DIM);
                    __builtin_nontemporal_store(u0,  ut + r0 * Q_DIM);
                    __builtin_nontemporal_store(u1,  ut + r1 * Q_DIM);
                }
            }
        }
        __syncthreads();
    }
}

extern "C" void kernel_launch(void* const* d_in, const int* in_sizes, int n_in,
                              void* d_out, int out_size, void* d_ws, size_t ws_size,
                              hipStream_t stream) {
    const float* x0 = (const float*)d_in[0];
    const float* S  = (const float*)d_in[1];
    const float* H1 = (const float*)d_in[2];
    const float* H2 = (const float*)d_in[3];
    float* xtraj = (float*)d_out;
    float* utraj = xtraj + (size_t)B_BATCH * (T_STEPS + 1) * N_NODES * P_DIM;

    const size_t shmem = (size_t)LDS_FLOATS * sizeof(float);  // ~99 KB / WGP

    (void)hipFuncSetAttribute((const void*)gcnn_rollout,
                              hipFuncAttributeMaxDynamicSharedMemorySize,
                              (int)shmem);

    gcnn_rollout<<<dim3(B_BATCH), dim3(THREADS), shmem, stream>>>(
        x0, S, H1, H2, xtraj, utraj);
}